// MorSP_4398046511104
// MI455X (gfx1250) — compile-verified
//
#include <hip/hip_runtime.h>
#include <stdint.h>

// MI455X / gfx1250 implementation of the skeleton fixed-point solver.
// Bandwidth-bound stencil workload: fused LDS-tiled separable 5-tap pool cascades
// with async-to-LDS copies; WMMA used for the energy-norm wave reduction.
//
// Scratch layout: 119 frames of 4MB (~476MB) in d_ws.

#define HW 512
#define NB 4
static const int NPIX_I = NB * HW * HW;               // 1,048,576
static const size_t NPIX = (size_t)NPIX_I;

#define NEG_BIG (-1e30f)

// ---------------- gfx1250 async-to-LDS helpers (CDNA5 ISA) ----------------

__device__ __forceinline__ uint32_t lds_off_of(const void* p) {
  // AS(3) -> flat: low 32 bits are the LDS byte offset (aperture is 4GB aligned).
  return (uint32_t)(uintptr_t)p;
}

__device__ __forceinline__ void async_ld_f32(uint32_t lds_off, const float* g) {
  unsigned long long ga = (unsigned long long)(uintptr_t)g;
  asm volatile("global_load_async_to_lds_b32 %0, %1, off"
               :: "v"(lds_off), "v"(ga) : "memory");
}

__device__ __forceinline__ void wait_async0() {
  asm volatile("s_wait_asynccnt 0" ::: "memory");
}

// ---------------- WMMA types --------------------------------------------------------

typedef __attribute__((ext_vector_type(2))) float v2f;
typedef __attribute__((ext_vector_type(8))) float v8f;

// ---------------- pointwise kernels -------------------------------------------------

__global__ __launch_bounds__(256) void k_init(const float* __restrict__ o,
                                              const float* __restrict__ v,
                                              float* __restrict__ u,
                                              float* __restrict__ q,
                                              float* __restrict__ w, int n) {
  int i = blockIdx.x * 256 + threadIdx.x;
  if (i >= n) return;
  float uu = 1.f / (1.f + __expf(-o[i]));
  float qq = fminf(fmaxf(v[i] - uu, -1.f), 1.f);
  u[i] = uu; q[i] = qq; w[i] = 0.5f * (uu + v[i]);
}

__global__ __launch_bounds__(256) void k_sub(const float* __restrict__ a,
                                             const float* __restrict__ b,
                                             float* __restrict__ d, int n) {
  int i = blockIdx.x * 256 + threadIdx.x;
  if (i >= n) return;
  d[i] = a[i] - b[i];
}

__global__ __launch_bounds__(256) void k_grad_init(const float* __restrict__ d,
                                                   const float* __restrict__ sumsq,
                                                   float* __restrict__ G, int n) {
  int i = blockIdx.x * 256 + threadIdx.x;
  if (i >= n) return;
  float s = fmaxf(*sumsq, 1e-30f);
  G[i] = d[i] * rsqrtf(s);              // d / ||d||
}

__global__ __launch_bounds__(256) void k_wupdate(float* __restrict__ w,
                                                 const float* __restrict__ q,
                                                 const float* __restrict__ g, int n) {
  int i = blockIdx.x * 256 + threadIdx.x;
  if (i >= n) return;
  w[i] = w[i] - 0.001f * (2.f * q[i] + 0.1f * g[i]);   // ITO*(ETA*q + DELTA*g)
}

__global__ __launch_bounds__(256) void k_final(const float* __restrict__ o,
                                               const float* __restrict__ p,
                                               const float* __restrict__ q,
                                               float* __restrict__ out, int n) {
  int i = blockIdx.x * 256 + threadIdx.x;
  if (i >= n) return;
  out[i] = o[i] - p[i] + 2.f * q[i];
}

// ---------------- fused forward skeleton step ---------------------------------------
// One launch per scan iteration t:
//   img_next = minpool5(img)          (valid-only SAME semantics via clamp)
//   o_t      = maxpool5(img_next)     (OOB tile cells masked to -BIG before max)
//   sk       = first ? relu(img - o) : skp + relu(relu(img-o)*(1-skp))
// 40x40 input tile (halo 4) -> 32x32 outputs. Async-to-LDS tile load.

__global__ __launch_bounds__(256) void k_skel_fwd(const float* __restrict__ img,
                                                  const float* __restrict__ skp,
                                                  float* __restrict__ img_next,
                                                  float* __restrict__ sk,
                                                  int first) {
  __shared__ float t0[40 * 40];   // img tile, halo 4
  __shared__ float t1[40 * 36];   // separable intermediate (reused)
  __shared__ float t2[36 * 36];   // E = img_next with halo 2 (masked)
  const int tx0 = blockIdx.x * 32, ty0 = blockIdx.y * 32, b = blockIdx.z;
  const size_t ib = (size_t)b * HW * HW;
  const uint32_t l0 = lds_off_of(t0);
  for (int e = threadIdx.x; e < 40 * 40; e += 256) {
    int lr = e / 40, lc = e - lr * 40;
    int cy = min(max(ty0 - 4 + lr, 0), HW - 1);
    int cx = min(max(tx0 - 4 + lc, 0), HW - 1);
    async_ld_f32(l0 + (uint32_t)e * 4u, img + ib + (size_t)cy * HW + cx);
  }
  wait_async0();
  __syncthreads();
  // horizontal min: t1[r][c], r in 0..39, c in 0..35
  for (int e = threadIdx.x; e < 40 * 36; e += 256) {
    int lr = e / 36, lc = e - lr * 36;
    const float* r = &t0[lr * 40 + lc];
    t1[e] = fminf(fminf(fminf(r[0], r[1]), fminf(r[2], r[3])), r[4]);
  }
  __syncthreads();
  // vertical min -> E (36x36); write img_next interior; mask OOB cells for max stage
  for (int e = threadIdx.x; e < 36 * 36; e += 256) {
    int lr = e / 36, lc = e - lr * 36;
    float m = t1[lr * 36 + lc];
#pragma unroll
    for (int k = 1; k < 5; k++) m = fminf(m, t1[(lr + k) * 36 + lc]);
    int gy = ty0 - 2 + lr, gx = tx0 - 2 + lc;
    if (lr >= 2 && lr < 34 && lc >= 2 && lc < 34)
      img_next[ib + (size_t)gy * HW + gx] = m;
    bool valid = (gy >= 0 && gy < HW && gx >= 0 && gx < HW);
    t2[e] = valid ? m : NEG_BIG;
  }
  __syncthreads();
  // horizontal max (reuse t1 as [36][32])
  for (int e = threadIdx.x; e < 36 * 32; e += 256) {
    int lr = e / 32, lc = e - lr * 32;
    const float* r = &t2[lr * 36 + lc];
    t1[e] = fmaxf(fmaxf(fmaxf(r[0], r[1]), fmaxf(r[2], r[3])), r[4]);
  }
  __syncthreads();
  // vertical max -> o_t; fused pointwise skel update
  const int tx = threadIdx.x & 31, ty = threadIdx.x >> 5;
  for (int s = 0; s < 4; s++) {
    int rr = ty + s * 8;
    float O = t1[rr * 32 + tx];
#pragma unroll
    for (int k = 1; k < 5; k++) O = fmaxf(O, t1[(rr + k) * 32 + tx]);
    size_t gi = ib + (size_t)(ty0 + rr) * HW + (tx0 + tx);
    float center = t0[(rr + 4) * 40 + (tx + 4)];
    float dl = fmaxf(center - O, 0.f);
    if (first) {
      sk[gi] = dl;
    } else {
      float sv = skp[gi];
      sk[gi] = sv + fmaxf(dl * (1.f - sv), 0.f);
    }
  }
}

// ---------------- fused backward stage 1 --------------------------------------------
// Per reverse step t:
//   recompute o_t = maxpool5(img_n) in-tile (bit-exact: min/max are rounding-free)
//   pointwise step backward ON THE HALO (36x36): Go, and for interior: gdir, Gout
//   fused maxpool-backward gather: tC[x] = gacc[x] + sum_y Go[y]*(img_n[x]==o_t[y])
// G is ping-ponged (Gin read on halo, Gout written interior) to avoid races.

__global__ __launch_bounds__(256) void k_skel_bwd1(const float* __restrict__ Gin,
                                                   float* __restrict__ Gout,
                                                   const float* __restrict__ img_t,
                                                   const float* __restrict__ img_n,
                                                   const float* __restrict__ skp,
                                                   const float* __restrict__ gacc,
                                                   float* __restrict__ gdir,
                                                   float* __restrict__ tC,
                                                   int first) {
  __shared__ float t0[40 * 40];   // img_n tile, halo 4
  __shared__ float t1[40 * 36];   // separable intermediate
  __shared__ float t2[36 * 36];   // o_t on halo 2
  __shared__ float t3[36 * 36];   // Go on halo 2 (0 where OOB)
  const int tx0 = blockIdx.x * 32, ty0 = blockIdx.y * 32, b = blockIdx.z;
  const size_t ib = (size_t)b * HW * HW;
  const uint32_t l0 = lds_off_of(t0);
  for (int e = threadIdx.x; e < 40 * 40; e += 256) {
    int lr = e / 40, lc = e - lr * 40;
    int cy = min(max(ty0 - 4 + lr, 0), HW - 1);
    int cx = min(max(tx0 - 4 + lc, 0), HW - 1);
    async_ld_f32(l0 + (uint32_t)e * 4u, img_n + ib + (size_t)cy * HW + cx);
  }
  wait_async0();
  __syncthreads();
  for (int e = threadIdx.x; e < 40 * 36; e += 256) {     // horizontal max
    int lr = e / 36, lc = e - lr * 36;
    const float* r = &t0[lr * 40 + lc];
    t1[e] = fmaxf(fmaxf(fmaxf(r[0], r[1]), fmaxf(r[2], r[3])), r[4]);
  }
  __syncthreads();
  for (int e = threadIdx.x; e < 36 * 36; e += 256) {     // vertical max -> o_t
    int lr = e / 36, lc = e - lr * 36;
    float m = t1[lr * 36 + lc];
#pragma unroll
    for (int k = 1; k < 5; k++) m = fmaxf(m, t1[(lr + k) * 36 + lc]);
    t2[e] = m;                       // clamp is exact for single-stage max
  }
  __syncthreads();
  // pointwise backward on the halo; write Gout/gdir for interior cells
  for (int e = threadIdx.x; e < 36 * 36; e += 256) {
    int lr = e / 36, lc = e - lr * 36;
    int gy = ty0 - 2 + lr, gx = tx0 - 2 + lc;
    float go = 0.f;
    if (gy >= 0 && gy < HW && gx >= 0 && gx < HW) {
      size_t g = ib + (size_t)gy * HW + gx;
      float gg = Gin[g];
      float diff = img_t[g] - t2[e];
      float dl = fmaxf(diff, 0.f);
      float gd, gnew = gg;
      if (first) {
        gd = gg;
      } else {
        float sv = skp[g];
        float aa = dl * (1.f - sv);
        if (aa > 0.f) { gd = gg * (1.f - sv); gnew = gg * (1.f - dl); }
        else          { gd = 0.f; }
      }
      float gi2 = (diff > 0.f) ? gd : 0.f;
      go = -gi2;
      if (lr >= 2 && lr < 34 && lc >= 2 && lc < 34) {
        gdir[g] = gi2;
        if (!first) Gout[g] = gnew;
      }
    }
    t3[e] = go;
  }
  __syncthreads();
  // fused maxpool-backward gather
  const int tx = threadIdx.x & 31, ty = threadIdx.x >> 5;
  for (int s = 0; s < 4; s++) {
    int rr = ty + s * 8;
    size_t gi = ib + (size_t)(ty0 + rr) * HW + (tx0 + tx);
    float x = t0[(rr + 4) * 40 + (tx + 4)];   // img_n center
    float acc = gacc[gi];
#pragma unroll
    for (int dy = 0; dy < 5; dy++)
#pragma unroll
      for (int dx = 0; dx < 5; dx++) {
        int li = (rr + dy) * 36 + (tx + dx);
        acc += (x == t2[li]) ? t3[li] : 0.f;
      }
    tC[gi] = acc;
  }
}

// ---------------- pool backward gather (minpool stage, B2) --------------------------
// gin[x] = extra[x] + sum_{y in N5(x)} gout[y]*(in[x]==pooled[y]).

__global__ __launch_bounds__(256) void k_pool5_bwd(const float* __restrict__ gout,
                                                   const float* __restrict__ in,
                                                   const float* __restrict__ pooled,
                                                   const float* __restrict__ extra,
                                                   float* __restrict__ gin) {
  __shared__ float tg[36 * 36];
  __shared__ float tp[36 * 36];
  const int tx0 = blockIdx.x * 32, ty0 = blockIdx.y * 32, b = blockIdx.z;
  const size_t ib = (size_t)b * HW * HW;
  for (int e = threadIdx.x; e < 36 * 36; e += 256) tg[e] = 0.f;
  __syncthreads();
  const uint32_t lg = lds_off_of(tg), lp = lds_off_of(tp);
  for (int e = threadIdx.x; e < 36 * 36; e += 256) {
    int lr = e / 36, lc = e - lr * 36;
    int gy = ty0 - 2 + lr, gx = tx0 - 2 + lc;
    int cy = min(max(gy, 0), HW - 1);
    int cx = min(max(gx, 0), HW - 1);
    async_ld_f32(lp + (uint32_t)e * 4u, pooled + ib + (size_t)cy * HW + cx);
    if (gy >= 0 && gy < HW && gx >= 0 && gx < HW)
      async_ld_f32(lg + (uint32_t)e * 4u, gout + ib + (size_t)gy * HW + gx);
  }
  wait_async0();
  __syncthreads();
  const int tx = threadIdx.x & 31, ty = threadIdx.x >> 5;
  for (int s = 0; s < 4; s++) {
    int rr = ty + s * 8;
    size_t gi = ib + (size_t)(ty0 + rr) * HW + (tx0 + tx);
    float x = in[gi];
    float acc = extra[gi];
#pragma unroll
    for (int dy = 0; dy < 5; dy++)
#pragma unroll
      for (int dx = 0; dx < 5; dx++) {
        int li = (rr + dy) * 36 + (tx + dx);
        acc += (x == tp[li]) ? tg[li] : 0.f;
      }
    gin[gi] = acc;
  }
}

// ---------------- fused outer iteration: p = 1 - 2*gauss5(u); u,q update -----------
// conv(1-2u, k) with zero pad == 1 - 2*conv(u with 0.5 pad) since sum(k)=1.

__global__ __launch_bounds__(256) void k_outer(const float* __restrict__ o,
                                               const float* __restrict__ uin,
                                               float* __restrict__ q,
                                               const float* __restrict__ w,
                                               float* __restrict__ p,
                                               float* __restrict__ uout) {
  __shared__ float t0[36 * 36];
  __shared__ float t1[36 * 32];
  const int tx0 = blockIdx.x * 32, ty0 = blockIdx.y * 32, b = blockIdx.z;
  const size_t ib = (size_t)b * HW * HW;
  for (int e = threadIdx.x; e < 36 * 36; e += 256) t0[e] = 0.5f;   // pad value
  __syncthreads();
  const uint32_t l0 = lds_off_of(t0);
  for (int e = threadIdx.x; e < 36 * 36; e += 256) {
    int lr = e / 36, lc = e - lr * 36;
    int gy = ty0 - 2 + lr, gx = tx0 - 2 + lc;
    if (gy >= 0 && gy < HW && gx >= 0 && gx < HW)
      async_ld_f32(l0 + (uint32_t)e * 4u, uin + ib + (size_t)gy * HW + gx);
  }
  wait_async0();
  __syncthreads();
  // separable normalized Gaussian, sigma=5, radius 2
  const float K0 = 0.19205110f, K1 = 0.20392650f, K2 = 0.20804480f;
  for (int e = threadIdx.x; e < 36 * 32; e += 256) {
    int lr = e / 32, lc = e - lr * 32;
    const float* r = &t0[lr * 36 + lc];
    t1[e] = K0 * (r[0] + r[4]) + K1 * (r[1] + r[3]) + K2 * r[2];
  }
  __syncthreads();
  const int tx = threadIdx.x & 31, ty = threadIdx.x >> 5;
  for (int s = 0; s < 4; s++) {
    int rr = ty + s * 8;
    float cv = K0 * (t1[rr * 32 + tx] + t1[(rr + 4) * 32 + tx]) +
               K1 * (t1[(rr + 1) * 32 + tx] + t1[(rr + 3) * 32 + tx]) +
               K2 * t1[(rr + 2) * 32 + tx];
    size_t gi = ib + (size_t)(ty0 + rr) * HW + (tx0 + tx);
    float pp = 1.f - 2.f * cv;
    float qq = q[gi];
    float un = 1.f / (1.f + __expf(-(o[gi] - pp + 2.f * qq)));
    float qn = fminf(fmaxf(qq + w[gi] - un, -1.f), 1.f);
    p[gi] = pp; uout[gi] = un; q[gi] = qn;
  }
}

// ---------------- WMMA norm reduction: sum(d^2) (deterministic, 2-stage) -----------

__global__ __launch_bounds__(256) void k_norm2_wmma(const float* __restrict__ d,
                                                    float* __restrict__ partial, int n) {
  float acc = 0.f;
  const int stride = gridDim.x * 256;
  for (int i = blockIdx.x * 256 + threadIdx.x; i < n; i += stride) {
    float x = d[i];
    acc += x * x;
  }
  v2f a; a.x = acc; a.y = 0.f;       // A[m,{0|2}] = lane partial, A[m,{1|3}] = 0
  v2f bb; bb.x = 1.f; bb.y = 1.f;    // B = ones
  v8f c = {};
  c = __builtin_amdgcn_wmma_f32_16x16x4_f32(false, a, false, bb, (short)0, c,
                                            false, false);
  float s = c[0] + c[1] + c[2] + c[3] + c[4] + c[5] + c[6] + c[7];
  float tot = s + __shfl_xor(s, 16, 32);   // rows 0-7 half + rows 8-15 half
  __shared__ float wsum[8];
  const int wave = threadIdx.x >> 5, lane = threadIdx.x & 31;
  if (lane == 0) wsum[wave] = tot;
  __syncthreads();
  if (threadIdx.x == 0) {
    float bs = 0.f;
    for (int i = 0; i < 8; i++) bs += wsum[i];
    partial[blockIdx.x] = bs;
  }
}

__global__ __launch_bounds__(64) void k_sum_final(const float* __restrict__ partial,
                                                  float* __restrict__ sumsq) {
  __shared__ float tmp[64];
  tmp[threadIdx.x] = partial[threadIdx.x];
  __syncthreads();
  if (threadIdx.x == 0) {
    float s = 0.f;
    for (int i = 0; i < 64; i++) s += tmp[i];
    *sumsq = s;
  }
}

// ---------------- host orchestration ------------------------------------------------

extern "C" void kernel_launch(void* const* d_in, const int* in_sizes, int n_in,
                              void* d_out, int out_size, void* d_ws, size_t ws_size,
                              hipStream_t stream) {
  (void)in_sizes; (void)n_in; (void)out_size; (void)ws_size;
  const float* o = (const float*)d_in[0];
  const float* v = (const float*)d_in[1];
  float* out = (float*)d_out;
  const int n = NPIX_I;
  const size_t FR = NPIX * sizeof(float);
  auto F = [&](int i) { return (float*)((char*)d_ws + (size_t)i * FR); };

  float *u0 = F(0), *u1 = F(1), *q = F(2), *w = F(3), *p = F(4), *skv = F(5),
        *dd = F(6), *Ga = F(7), *Gb = F(8), *gA = F(9), *gB = F(10), *tA = F(11),
        *tC = F(12), *vA = F(13), *vB = F(14);
  float* imgs[52]; for (int t = 0; t < 52; t++) imgs[t] = F(15 + t);
  float* sks[51];  for (int t = 0; t < 51; t++) sks[t] = F(67 + t);
  float* partial = F(118);          // 64 per-block partials
  float* sumsq = partial + 64;      // scalar

  dim3 PB(256), PG((unsigned)((n + 255) / 256));
  dim3 TB(256), TG(HW / 32, HW / 32, NB);

  // init u, q, w
  k_init<<<PG, PB, 0, stream>>>(o, v, u0, q, w, n);

  // skel(v): computed once, cached (v is constant across egrad calls)
  hipMemcpyAsync(vA, v, FR, hipMemcpyDeviceToDevice, stream);
  {
    float *cur = vA, *nxt = vB;
    for (int t = 0; t <= 50; t++) {
      k_skel_fwd<<<TG, TB, 0, stream>>>(cur, skv, nxt, skv, (int)(t == 0));
      float* tmp = cur; cur = nxt; nxt = tmp;
    }
  }

  float *uc = u0, *un = u1;
  for (int i = 0; i < 20; i++) {
    // p = conv(1-2u); u = sigmoid(o - p + 2q); q = clip(q + w - u)
    k_outer<<<TG, TB, 0, stream>>>(o, uc, q, w, p, un);
    { float* t = uc; uc = un; un = t; }

    if (i % 5 == 0) {
      // ---- forward skel(w) with recording (img[t+1] = minpool(img[t]) = e_t) ----
      hipMemcpyAsync(imgs[0], w, FR, hipMemcpyDeviceToDevice, stream);
      for (int t = 0; t <= 50; t++)
        k_skel_fwd<<<TG, TB, 0, stream>>>(imgs[t], t ? sks[t - 1] : sks[0],
                                          imgs[t + 1], sks[t], (int)(t == 0));
      // ---- energy norm + seed gradient: G = d / ||d|| ----
      k_sub<<<PG, PB, 0, stream>>>(sks[50], skv, dd, n);
      k_norm2_wmma<<<dim3(64), PB, 0, stream>>>(dd, partial, n);
      k_sum_final<<<dim3(1), dim3(64), 0, stream>>>(partial, sumsq);
      k_grad_init<<<PG, PB, 0, stream>>>(dd, sumsq, Ga, n);
      // ---- reverse scan ----
      hipMemsetAsync(gA, 0, FR, stream);                    // dE/d img[51]
      float *Gc = Ga, *Gn = Gb;
      float *gc = gA, *gn = gB;
      for (int t = 50; t >= 0; t--) {
        // B1: o_t recompute + pointwise bwd + maxpool-bwd gather (fused)
        k_skel_bwd1<<<TG, TB, 0, stream>>>(Gc, Gn, imgs[t], imgs[t + 1],
                                           t ? sks[t - 1] : sks[0], gc, tA, tC,
                                           (int)(t == 0));
        { float* tt = Gc; Gc = Gn; Gn = tt; }
        // B2: minpool-bwd gather + direct delta path
        k_pool5_bwd<<<TG, TB, 0, stream>>>(tC, imgs[t], imgs[t + 1], tA, gn);
        { float* tt = gc; gc = gn; gn = tt; }
      }
      // gc == dE/dw
      k_wupdate<<<PG, PB, 0, stream>>>(w, q, gc, n);
    }
  }

  k_final<<<PG, PB, 0, stream>>>(o, p, q, out, n);
}